// ModelA_40192303956383
// MI455X (gfx1250) — compile-verified
//
#include <hip/hip_runtime.h>
#include <hip/hip_bf16.h>

typedef __attribute__((ext_vector_type(2))) float v2f;
typedef __attribute__((ext_vector_type(8))) float v8f;

#define N_NODES 50000
#define N_EDGES 800000
#define DIM 128
#define FEAT3 384
#define H2DIM 64

// ---------------------------------------------------------------------------
// Zero-fill
// ---------------------------------------------------------------------------
__global__ void gcn_zero_kernel(float* __restrict__ p, long n) {
    long i = (long)blockIdx.x * blockDim.x + threadIdx.x;
    if (i < n) p[i] = 0.0f;
}

// ---------------------------------------------------------------------------
// Tiled fp32 WMMA GEMM:  C[M x (NT*16)] = A[M x K] @ B[K x (NT*16)] (+bias)(relu)
//   - block = 256 threads = 8 waves; each wave computes a 16-row stripe
//   - B chunk (32 x NT*16) double-buffered in LDS; the NEXT chunk is
//     prefetched with CDNA5 async-to-LDS ops (global_load_async_to_lds_b128,
//     tracked by ASYNCcnt) while WMMAs consume the current one.
//   - REQUIRES ldb == NT*16 (true at every call site) so each 32-row B chunk
//     is a contiguous span of B -> perfectly coalesced b128 async copies.
//   - uses V_WMMA_F32_16X16X4_F32 (fp32 matrix core, exact precision)
// A fragment (ISA 7.12.2, 32-bit A 16x4): lanes 0-15 hold K=k0,k0+1,
// lanes 16-31 hold K=k0+2,k0+3, row M = lane&15.
// B fragment: lanes 0-15 VGPR0=K0 row, VGPR1=K1 row; lanes 16-31 VGPR0=K2,
// VGPR1=K3. C/D: VGPR j -> M=j (lanes 0-15) / M=j+8 (lanes 16-31), N=lane&15.
// ---------------------------------------------------------------------------
template <int NT>
__global__ __launch_bounds__(256) void gcn_gemm_wmma(
    const float* __restrict__ A, int lda,
    const float* __restrict__ B, int /*ldb == NT*16*/,
    float* __restrict__ C, int ldc,
    int M, int K,
    const float* __restrict__ bias, int do_relu) {

    constexpr int NCOLS = NT * 16;
    constexpr int CHUNK = 32 * NCOLS;            // floats per K-chunk
    constexpr int NB128 = CHUNK / 4 / 256;       // b128 async loads per thread
    __shared__ float Bs[2][CHUNK];

    const int tid = threadIdx.x;
    const int lane = tid & 31;
    const int wave = tid >> 5;
    const int r0 = blockIdx.x * 128 + wave * 16;
    const int mr = r0 + (lane & 15);          // A-fragment row for this lane
    const int khalf = (lane >> 4) << 1;       // 0 for lanes 0-15, 2 for 16-31

    // async prefetch of one 32 x NCOLS chunk of B into Bs[buf]
    auto stage = [&](int kc, int buf) {
#pragma unroll
        for (int j = 0; j < NB128; ++j) {
            int idx = tid + j * 256;                               // b128 index
            unsigned ldsa = (unsigned)(uintptr_t)&Bs[buf][idx * 4]; // LDS byte addr
            unsigned goff = (unsigned)((kc * NCOLS + idx * 4) * sizeof(float));
            asm volatile("global_load_async_to_lds_b128 %0, %1, %2"
                         :: "v"(ldsa), "v"(goff), "s"(B)
                         : "memory");
        }
    };

    v8f acc[NT];
#pragma unroll
    for (int t = 0; t < NT; ++t) acc[t] = (v8f){0.f, 0.f, 0.f, 0.f, 0.f, 0.f, 0.f, 0.f};

    // prologue: stage chunk 0
    stage(0, 0);
    asm volatile("s_wait_asynccnt 0x0" ::: "memory");
    __syncthreads();

    int buf = 0;
    for (int kc = 0; kc < K; kc += 32) {
        // prefetch next chunk into the other buffer while computing this one
        if (kc + 32 < K) stage(kc + 32, buf ^ 1);

#pragma unroll
        for (int kk = 0; kk < 32; kk += 4) {
            // A fragment (16x4 f32)
            v2f a;
            if (mr < M) {
                const float* ap = A + (size_t)mr * lda + (kc + kk + khalf);
                a.x = ap[0];
                a.y = ap[1];
            } else {
                a.x = 0.f;
                a.y = 0.f;
            }
            // hoist all B fragments for this k-step, then WMMA burst
            v2f bf[NT];
#pragma unroll
            for (int t = 0; t < NT; ++t) {
                int ncol = t * 16 + (lane & 15);
                bf[t].x = Bs[buf][(kk + khalf) * NCOLS + ncol];
                bf[t].y = Bs[buf][(kk + khalf + 1) * NCOLS + ncol];
            }
#pragma unroll
            for (int t = 0; t < NT; ++t) {
                acc[t] = __builtin_amdgcn_wmma_f32_16x16x4_f32(
                    false, a, false, bf[t], (short)0, acc[t], false, false);
            }
        }

        // my async writes for the next buffer are done; barrier makes them
        // visible block-wide and confirms everyone finished reading Bs[buf]
        asm volatile("s_wait_asynccnt 0x0" ::: "memory");
        __syncthreads();
        buf ^= 1;
    }

    // epilogue: optional bias + relu, store
#pragma unroll
    for (int t = 0; t < NT; ++t) {
        int ncol = t * 16 + (lane & 15);
        float bv = bias ? bias[ncol] : 0.0f;
#pragma unroll
        for (int j = 0; j < 8; ++j) {
            int m = r0 + j + ((lane >> 4) << 3);
            if (m < M) {
                float v = acc[t][j] + bv;
                if (do_relu) v = fmaxf(v, 0.0f);
                C[(size_t)m * ldc + ncol] = v;
            }
        }
    }
}

// ---------------------------------------------------------------------------
// SpMM scatter: agg[row[e]*384 + off + f] += vals[e] * sup[col[e]*128 + f]
// One thread per (edge, feature); lanes cover 32 consecutive features of one
// edge -> coalesced gather; hardware fp32 atomic scatter (L2-side).
// ---------------------------------------------------------------------------
__global__ void gcn_spmm_atomic(const int* __restrict__ row,
                                const int* __restrict__ col,
                                const float* __restrict__ vals,
                                const float* __restrict__ sup,
                                float* __restrict__ agg, int off) {
    size_t tid = (size_t)blockIdx.x * blockDim.x + threadIdx.x;
    if (tid >= (size_t)N_EDGES * DIM) return;
    int e = (int)(tid >> 7);
    int f = (int)(tid & 127);
    int r = row[e];
    int c = col[e];
    float v = vals[e] * sup[(size_t)c * DIM + f];
    unsafeAtomicAdd(&agg[(size_t)r * FEAT3 + off + f], v);
}

// ---------------------------------------------------------------------------
// Elementwise bias + relu on a 128-wide stripe of the strided feat buffer
// ---------------------------------------------------------------------------
__global__ void gcn_bias_relu(float* __restrict__ feat,
                              const float* __restrict__ b, int off) {
    size_t tid = (size_t)blockIdx.x * blockDim.x + threadIdx.x;
    if (tid >= (size_t)N_NODES * DIM) return;
    int r = (int)(tid >> 7);
    int f = (int)(tid & 127);
    float* p = &feat[(size_t)r * FEAT3 + off + f];
    *p = fmaxf(*p + b[f], 0.0f);
}

// ---------------------------------------------------------------------------
// Final tiny FC: logits[N x 2] = h2[N x 64] @ fcW3[64 x 2] + fcb3
// ---------------------------------------------------------------------------
__global__ void gcn_fc3(const float* __restrict__ h2,
                        const float* __restrict__ W,
                        const float* __restrict__ b,
                        float* __restrict__ out) {
    int r = blockIdx.x * blockDim.x + threadIdx.x;
    if (r >= N_NODES) return;
    float a0 = b[0], a1 = b[1];
    const float* hp = h2 + (size_t)r * H2DIM;
#pragma unroll
    for (int j = 0; j < H2DIM; ++j) {
        float h = hp[j];
        a0 = fmaf(h, W[j * 2 + 0], a0);
        a1 = fmaf(h, W[j * 2 + 1], a1);
    }
    out[(size_t)r * 2 + 0] = a0;
    out[(size_t)r * 2 + 1] = a1;
}

// ---------------------------------------------------------------------------
// Orchestration
// ---------------------------------------------------------------------------
extern "C" void kernel_launch(void* const* d_in, const int* in_sizes, int n_in,
                              void* d_out, int out_size, void* d_ws, size_t ws_size,
                              hipStream_t stream) {
    const int*   adj_row = (const int*)d_in[0];
    const int*   adj_col = (const int*)d_in[1];
    const float* adj_val = (const float*)d_in[2];
    const float* X       = (const float*)d_in[3];
    // d_in[4] = graph_indicator (unused by reference)
    const float* W1 = (const float*)d_in[5];
    const float* b1 = (const float*)d_in[6];
    const float* W2 = (const float*)d_in[7];
    const float* b2 = (const float*)d_in[8];
    const float* W3 = (const float*)d_in[9];
    const float* b3 = (const float*)d_in[10];
    const float* fcW1 = (const float*)d_in[11];
    const float* fcb1 = (const float*)d_in[12];
    const float* fcW2 = (const float*)d_in[13];
    const float* fcb2 = (const float*)d_in[14];
    const float* fcW3 = (const float*)d_in[15];
    const float* fcb3 = (const float*)d_in[16];
    float* out = (float*)d_out;

    // workspace layout
    float* sup  = (float*)d_ws;                    // N x 128
    float* feat = sup  + (size_t)N_NODES * DIM;    // N x 384 (g1|g2|g3)
    float* h1   = feat + (size_t)N_NODES * FEAT3;  // N x 128
    float* h2   = h1   + (size_t)N_NODES * DIM;    // N x 64

    const int gemm_blocks = (N_NODES + 127) / 128;
    const int ew_n  = (int)(((size_t)N_NODES * DIM + 255) / 256);
    const int ew_e  = (int)(((size_t)N_EDGES * DIM + 255) / 256);
    const int zr_n  = (int)(((size_t)N_NODES * FEAT3 + 255) / 256);

    // zero the aggregation buffer (atomics accumulate into it)
    gcn_zero_kernel<<<zr_n, 256, 0, stream>>>(feat, (long)N_NODES * FEAT3);

    // ---- GCN layer 1 ----
    gcn_gemm_wmma<8><<<gemm_blocks, 256, 0, stream>>>(X, DIM, W1, DIM, sup, DIM,
                                                      N_NODES, DIM, nullptr, 0);
    gcn_spmm_atomic<<<ew_e, 256, 0, stream>>>(adj_row, adj_col, adj_val, sup, feat, 0);
    gcn_bias_relu<<<ew_n, 256, 0, stream>>>(feat, b1, 0);

    // ---- GCN layer 2 (input = g1 = feat[:,0:128], lda=384) ----
    gcn_gemm_wmma<8><<<gemm_blocks, 256, 0, stream>>>(feat + 0, FEAT3, W2, DIM, sup, DIM,
                                                      N_NODES, DIM, nullptr, 0);
    gcn_spmm_atomic<<<ew_e, 256, 0, stream>>>(adj_row, adj_col, adj_val, sup, feat, DIM);
    gcn_bias_relu<<<ew_n, 256, 0, stream>>>(feat, b2, DIM);

    // ---- GCN layer 3 (input = g2 = feat[:,128:256]) ----
    gcn_gemm_wmma<8><<<gemm_blocks, 256, 0, stream>>>(feat + DIM, FEAT3, W3, DIM, sup, DIM,
                                                      N_NODES, DIM, nullptr, 0);
    gcn_spmm_atomic<<<ew_e, 256, 0, stream>>>(adj_row, adj_col, adj_val, sup, feat, 2 * DIM);
    gcn_bias_relu<<<ew_n, 256, 0, stream>>>(feat, b3, 2 * DIM);

    // ---- FC head ----
    // h1 = relu(feat @ fcW1 + fcb1)   (K = 384)
    gcn_gemm_wmma<8><<<gemm_blocks, 256, 0, stream>>>(feat, FEAT3, fcW1, DIM, h1, DIM,
                                                      N_NODES, FEAT3, fcb1, 1);
    // h2 = relu(h1 @ fcW2 + fcb2)     (K = 128, Ncols = 64)
    gcn_gemm_wmma<4><<<gemm_blocks, 256, 0, stream>>>(h1, DIM, fcW2, H2DIM, h2, H2DIM,
                                                      N_NODES, DIM, fcb2, 1);
    // logits = h2 @ fcW3 + fcb3
    gcn_fc3<<<(N_NODES + 255) / 256, 256, 0, stream>>>(h2, fcW3, fcb3, out);
}